// pointnet_fp_module_24532853195517
// MI455X (gfx1250) — compile-verified
//
#include <hip/hip_runtime.h>
#include <hip/hip_bf16.h>

// ---------------- problem constants ----------------
#define BB   8
#define NN   16384
#define SS   2048
#define C1C  128
#define C2C  256
#define CIN0 384           // C1 + C2
#define COUT 256
#define BN_EPS 1e-5f

typedef __attribute__((ext_vector_type(2))) float v2f;
typedef __attribute__((ext_vector_type(8))) float v8f;

// ---------------- workspace layout (float units) ----------------
static const size_t OFF_P2T = 0;                                   // [B][S][C2]
static const size_t OFF_IDX = OFF_P2T + (size_t)BB * SS * C2C;     // [B][N][3] (int)
static const size_t OFF_WGT = OFF_IDX + (size_t)BB * NN * 3;       // [B][N][3]
static const size_t OFF_H0  = OFF_WGT + (size_t)BB * NN * 3;       // [B][384][N]  (reused as h1)
static const size_t OFF_Y0  = OFF_H0  + (size_t)BB * CIN0 * NN;    // [B][256][N]  (reused as y1)
static const size_t OFF_ST0 = OFF_Y0  + (size_t)BB * COUT * NN;    // sum[256], sumsq[256]
static const size_t OFF_ST1 = OFF_ST0 + 512;
static const size_t OFF_SS0 = OFF_ST1 + 512;                       // scale[256], shift[256]
static const size_t OFF_SS1 = OFF_SS0 + 512;

// ---------------- 1) transpose points2 (B,C2,S) -> (B,S,C2) ----------------
__global__ __launch_bounds__(256)
void k_transpose_p2(const float* __restrict__ p2, float* __restrict__ p2t) {
    const size_t total = (size_t)BB * C2C * SS;
    for (size_t i = (size_t)blockIdx.x * blockDim.x + threadIdx.x; i < total;
         i += (size_t)gridDim.x * blockDim.x) {
        size_t b = i / ((size_t)C2C * SS);
        size_t r = i % ((size_t)C2C * SS);
        size_t c = r / SS, s = r % SS;
        p2t[(b * SS + s) * C2C + c] = p2[i];
    }
}

// ---------------- 2) 3-NN search (xyz2 tile in LDS) ----------------
__global__ __launch_bounds__(256)
void k_knn3(const float* __restrict__ xyz1, const float* __restrict__ xyz2,
            int* __restrict__ idxb, float* __restrict__ wb) {
    __shared__ float sx[SS], sy[SS], sz[SS];
    const int b = blockIdx.y;
    const float* p2 = xyz2 + (size_t)b * SS * 3;
    for (int i = threadIdx.x; i < SS; i += 256) {
        sx[i] = p2[i * 3 + 0];
        sy[i] = p2[i * 3 + 1];
        sz[i] = p2[i * 3 + 2];
    }
    __syncthreads();

    const int n = blockIdx.x * 256 + threadIdx.x;
    const float* q = xyz1 + ((size_t)b * NN + n) * 3;
    const float qx = q[0], qy = q[1], qz = q[2];

    float d0 = 3.0e38f, d1 = 3.0e38f, d2 = 3.0e38f;
    int   i0 = 0, i1 = 0, i2 = 0;
    for (int s = 0; s < SS; ++s) {
        const float dx = qx - sx[s], dy = qy - sy[s], dz = qz - sz[s];
        const float d = dx * dx + dy * dy + dz * dz;
        if (d < d2) {
            if (d < d0)      { d2 = d1; i2 = i1; d1 = d0; i1 = i0; d0 = d; i0 = s; }
            else if (d < d1) { d2 = d1; i2 = i1; d1 = d;  i1 = s; }
            else             { d2 = d;  i2 = s; }
        }
    }
    d0 = fmaxf(d0, 1e-10f); d1 = fmaxf(d1, 1e-10f); d2 = fmaxf(d2, 1e-10f);
    float w0 = 1.0f / d0, w1 = 1.0f / d1, w2 = 1.0f / d2;
    const float inv = 1.0f / (w0 + w1 + w2);
    const size_t o = ((size_t)b * NN + n) * 3;
    idxb[o] = i0; idxb[o + 1] = i1; idxb[o + 2] = i2;
    wb[o] = w0 * inv; wb[o + 1] = w1 * inv; wb[o + 2] = w2 * inv;
}

// ---------------- 3) interpolate + concat -> h0 (B,384,N) ----------------
__global__ __launch_bounds__(256)
void k_interp_concat(const float* __restrict__ p2t, const int* __restrict__ idxb,
                     const float* __restrict__ wb, const float* __restrict__ points1,
                     float* __restrict__ h0) {
    const int b = blockIdx.y;
    const int n = blockIdx.x * 256 + threadIdx.x;
    const size_t o = ((size_t)b * NN + n) * 3;
    const int i0 = idxb[o], i1 = idxb[o + 1], i2 = idxb[o + 2];
    const float w0 = wb[o], w1 = wb[o + 1], w2 = wb[o + 2];
    const float4* r0 = (const float4*)(p2t + ((size_t)b * SS + i0) * C2C);
    const float4* r1 = (const float4*)(p2t + ((size_t)b * SS + i1) * C2C);
    const float4* r2 = (const float4*)(p2t + ((size_t)b * SS + i2) * C2C);
    float* hb = h0 + (size_t)b * CIN0 * NN;

    for (int c4 = 0; c4 < C2C / 4; ++c4) {
        const float4 a0 = r0[c4], a1 = r1[c4], a2 = r2[c4];
        const int c = c4 * 4;
        hb[(size_t)(c + 0) * NN + n] = w0 * a0.x + w1 * a1.x + w2 * a2.x;
        hb[(size_t)(c + 1) * NN + n] = w0 * a0.y + w1 * a1.y + w2 * a2.y;
        hb[(size_t)(c + 2) * NN + n] = w0 * a0.z + w1 * a1.z + w2 * a2.z;
        hb[(size_t)(c + 3) * NN + n] = w0 * a0.w + w1 * a1.w + w2 * a2.w;
    }
    const float* p1b = points1 + (size_t)b * C1C * NN;
    for (int c = 0; c < C1C; ++c)
        hb[(size_t)(C2C + c) * NN + n] = p1b[(size_t)c * NN + n];
}

// ---------------- 4) WMMA fp32 GEMM: Y[b,m,n] = sum_k W[m,k]*X[b,k,n] + bias[m] ----------------
// grid (N/16, B), block 512 = 16 waves; wave w owns output tile rows [16w,16w+16), cols [16*bx,...)
__global__ __launch_bounds__(512)
void k_gemm_wmma(const float* __restrict__ W, const float* __restrict__ bias,
                 const float* __restrict__ X, float* __restrict__ Y, int K) {
    // LDS tile of X: K x 16, packed [k>>1][n][k&1] so each lane's B-operand pair is contiguous (b64)
    __shared__ float lds[CIN0 * 16];
    const int b  = blockIdx.y;
    const int n0 = blockIdx.x * 16;
    const int t  = threadIdx.x;
    const float* Xb = X + (size_t)b * K * NN;

    for (int i = t; i < K * 16; i += 512) {
        const int k = i >> 4, n = i & 15;
        lds[(((k >> 1) * 16) + n) * 2 + (k & 1)] = Xb[(size_t)k * NN + n0 + n];
    }
    __syncthreads();

    const int wave = t >> 5, lane = t & 31;
    const int m0 = wave * 16;
    const int lhalf = lane >> 4, l15 = lane & 15;

    v8f acc = {};
    const float* Wrow = W + (size_t)(m0 + l15) * K;   // A: lane holds row m0+l15, K-pair 2*lhalf
    for (int k0 = 0; k0 < K; k0 += 4) {
        const int ka = k0 + 2 * lhalf;                // even
        const float2 aw = *(const float2*)(Wrow + ka);
        const float2 bw = *(const float2*)(&lds[(((ka >> 1) * 16) + l15) * 2]);
        v2f a; a.x = aw.x; a.y = aw.y;                // A[m][ka], A[m][ka+1]
        v2f bb; bb.x = bw.x; bb.y = bw.y;             // X[ka][n], X[ka+1][n]
        acc = __builtin_amdgcn_wmma_f32_16x16x4_f32(
            /*neg_a=*/false, a, /*neg_b=*/false, bb,
            /*c_mod=*/(short)0, acc, /*reuse_a=*/false, /*reuse_b=*/false);
    }

    // D layout: vgpr r -> M = r + 8*lhalf, N = l15
    float* Yb = Y + ((size_t)b * COUT + m0) * NN;
    for (int r = 0; r < 8; ++r) {
        const int m = r + 8 * lhalf;
        Yb[(size_t)m * NN + n0 + l15] = acc[r] + bias[m0 + m];
    }
}

// ---------------- 5) batchnorm statistics (per-channel sum / sumsq) ----------------
__global__ __launch_bounds__(512)
void k_zero_stats(float* s0, float* s1) {
    const int t = blockIdx.x * blockDim.x + threadIdx.x;
    if (t < 512) { s0[t] = 0.0f; s1[t] = 0.0f; }
}

__global__ __launch_bounds__(256)
void k_bn_stats(const float* __restrict__ Y, float* __restrict__ stats) {
    __shared__ float ssum[256], ssq[256];
    const int m = blockIdx.x;
    const int base = blockIdx.y * 2048;               // flat index into B*N
    float s = 0.0f, q = 0.0f;
    for (int i = threadIdx.x; i < 2048; i += 256) {
        const int f = base + i;
        const int b = f >> 14, n = f & (NN - 1);
        const float v = Y[((size_t)b * COUT + m) * NN + n];
        s += v; q += v * v;
    }
    ssum[threadIdx.x] = s; ssq[threadIdx.x] = q;
    __syncthreads();
    for (int off = 128; off > 0; off >>= 1) {
        if ((int)threadIdx.x < off) {
            ssum[threadIdx.x] += ssum[threadIdx.x + off];
            ssq[threadIdx.x]  += ssq[threadIdx.x + off];
        }
        __syncthreads();
    }
    if (threadIdx.x == 0) {
        atomicAdd(&stats[m], ssum[0]);
        atomicAdd(&stats[COUT + m], ssq[0]);
    }
}

__global__ __launch_bounds__(256)
void k_bn_finalize(const float* __restrict__ stats, const float* __restrict__ g,
                   const float* __restrict__ beta, float* __restrict__ scshift) {
    const int m = threadIdx.x;
    const float inv = 1.0f / ((float)BB * (float)NN);
    const float mean = stats[m] * inv;
    const float var  = stats[COUT + m] * inv - mean * mean;
    const float sc   = g[m] * rsqrtf(var + BN_EPS);
    scshift[m] = sc;
    scshift[COUT + m] = beta[m] - mean * sc;
}

// ---------------- 6) fused BN apply + ReLU ----------------
__global__ __launch_bounds__(256)
void k_bn_relu(const float* __restrict__ Y, const float* __restrict__ scshift,
               float* __restrict__ out) {
    const size_t total4 = (size_t)BB * COUT * NN / 4;
    for (size_t i = (size_t)blockIdx.x * blockDim.x + threadIdx.x; i < total4;
         i += (size_t)gridDim.x * blockDim.x) {
        const int m = (int)((i * 4 / NN) % COUT);
        const float sc = scshift[m], sh = scshift[COUT + m];
        const float4 y = ((const float4*)Y)[i];
        float4 v;
        v.x = fmaxf(y.x * sc + sh, 0.0f);
        v.y = fmaxf(y.y * sc + sh, 0.0f);
        v.z = fmaxf(y.z * sc + sh, 0.0f);
        v.w = fmaxf(y.w * sc + sh, 0.0f);
        ((float4*)out)[i] = v;
    }
}

// ---------------- launcher ----------------
extern "C" void kernel_launch(void* const* d_in, const int* in_sizes, int n_in,
                              void* d_out, int out_size, void* d_ws, size_t ws_size,
                              hipStream_t stream) {
    const float* xyz1    = (const float*)d_in[0];
    const float* xyz2    = (const float*)d_in[1];
    const float* points1 = (const float*)d_in[2];
    const float* points2 = (const float*)d_in[3];
    const float* W0 = (const float*)d_in[4];
    const float* b0 = (const float*)d_in[5];
    const float* g0 = (const float*)d_in[6];
    const float* be0 = (const float*)d_in[7];
    const float* W1 = (const float*)d_in[8];
    const float* b1 = (const float*)d_in[9];
    const float* g1 = (const float*)d_in[10];
    const float* be1 = (const float*)d_in[11];

    float* ws   = (float*)d_ws;
    float* p2t  = ws + OFF_P2T;
    int*   idxb = (int*)(ws + OFF_IDX);
    float* wb   = ws + OFF_WGT;
    float* h0   = ws + OFF_H0;      // also h1
    float* y0   = ws + OFF_Y0;      // also y1
    float* st0  = ws + OFF_ST0;
    float* st1  = ws + OFF_ST1;
    float* ss0  = ws + OFF_SS0;
    float* ss1  = ws + OFF_SS1;
    float* outp = (float*)d_out;

    // 1) transpose points2 for contiguous gathers
    k_transpose_p2<<<4096, 256, 0, stream>>>(points2, p2t);
    // 2) 3-NN
    k_knn3<<<dim3(NN / 256, BB), 256, 0, stream>>>(xyz1, xyz2, idxb, wb);
    // 3) interpolate + concat -> h0
    k_interp_concat<<<dim3(NN / 256, BB), 256, 0, stream>>>(p2t, idxb, wb, points1, h0);
    // stats buffers must be zero every call (graph replay safe)
    k_zero_stats<<<1, 512, 0, stream>>>(st0, st1);

    // layer 0: GEMM (K=384) -> y0, BN stats, BN+ReLU -> h1 (reuses h0 region)
    k_gemm_wmma<<<dim3(NN / 16, BB), 512, 0, stream>>>(W0, b0, h0, y0, CIN0);
    k_bn_stats<<<dim3(COUT, (BB * NN) / 2048), 256, 0, stream>>>(y0, st0);
    k_bn_finalize<<<1, 256, 0, stream>>>(st0, g0, be0, ss0);
    k_bn_relu<<<8192, 256, 0, stream>>>(y0, ss0, h0);   // h1 overwrites h0 region

    // layer 1: GEMM (K=256) -> y1 (reuses y0 region), BN stats, BN+ReLU -> d_out
    k_gemm_wmma<<<dim3(NN / 16, BB), 512, 0, stream>>>(W1, b1, h0, y0, COUT);
    k_bn_stats<<<dim3(COUT, (BB * NN) / 2048), 256, 0, stream>>>(y0, st1);
    k_bn_finalize<<<1, 256, 0, stream>>>(st1, g1, be1, ss1);
    k_bn_relu<<<8192, 256, 0, stream>>>(y0, ss1, outp);
}